// GAT_14285061226769
// MI455X (gfx1250) — compile-verified
//
#include <hip/hip_runtime.h>
#include <math.h>

typedef float v2f __attribute__((ext_vector_type(2)));
typedef float v8f __attribute__((ext_vector_type(8)));

#define NEG_SLOPE 0.2f
#define EPSF 1e-16f

// ---- ordered-float <-> uint mapping so atomicMax(uint) == float max ----
__device__ __forceinline__ unsigned f2ord(float f) {
  unsigned u = __float_as_uint(f);
  return (u & 0x80000000u) ? ~u : (u | 0x80000000u);
}
__device__ __forceinline__ float ord2f(unsigned u) {
  return __uint_as_float((u & 0x80000000u) ? (u ^ 0x80000000u) : ~u);
}

// CDNA5 async copy: global -> LDS, 16B per lane, tracked by ASYNCcnt.
// lds_off: LDS byte offset (low 32 bits of flat LDS address per ISA §10.2)
__device__ __forceinline__ void async_g2l_b128(unsigned lds_off,
                                               const void* gaddr) {
  asm volatile("global_load_async_to_lds_b128 %0, %1, off"
               :
               : "v"(lds_off), "v"((unsigned long long)(uintptr_t)gaddr)
               : "memory");
}
__device__ __forceinline__ void wait_async0() {
  asm volatile("s_wait_asynccnt 0" ::: "memory");
}

__global__ void zero_kernel(float* __restrict__ p, long long n) {
  long long i = (long long)blockIdx.x * blockDim.x + threadIdx.x;
  long long stride = (long long)gridDim.x * blockDim.x;
  for (; i < n; i += stride) p[i] = 0.0f;
}

// ------------------------------------------------------------------
// GEMM1: H[N,128] = X[N,128] * W[128,128]   (f32 WMMA 16x16x4)
// W async-staged to LDS, then repacked so each lane's K-pair is one
// aligned b64: (k,n) -> lds[((k>>1)*128 + n)*2 + (k&1)]
// ------------------------------------------------------------------
__global__ __launch_bounds__(256) void gemm1_kernel(
    const float* __restrict__ X, const float* __restrict__ W,
    float* __restrict__ Hout, int N) {
  __shared__ float ldsW[128 * 128];
  // 1) async copy W verbatim: 4096 x b128 chunks, 16 per thread
  {
    unsigned base = (unsigned)(uintptr_t)ldsW;
    for (int i = threadIdx.x; i < 128 * 128 / 4; i += 256)
      async_g2l_b128(base + i * 16u, W + i * 4);
    wait_async0();
  }
  __syncthreads();
  // 2) repack in place: 64 floats per thread through registers
  {
    float tmp[64];
#pragma unroll
    for (int j = 0; j < 64; ++j) tmp[j] = ldsW[threadIdx.x * 64 + j];
    __syncthreads();
#pragma unroll
    for (int j = 0; j < 64; ++j) {
      int idx = threadIdx.x * 64 + j;
      int k = idx >> 7, n = idx & 127;
      ldsW[((k >> 1) * 128 + n) * 2 + (k & 1)] = tmp[j];
    }
  }
  __syncthreads();

  int wave = threadIdx.x >> 5;
  int lane = threadIdx.x & 31;
  int tile = blockIdx.x * 8 + wave;
  if (tile * 16 >= N) return;
  int m = lane & 15, kh = lane >> 4;
  long long row = (long long)tile * 16 + m;

  // A-tile: each lane's full K footprint (shared across all 8 col tiles)
  float areg[64];
#pragma unroll
  for (int s = 0; s < 32; ++s) {
    areg[2 * s]     = X[row * 128 + 4 * s + 2 * kh];
    areg[2 * s + 1] = X[row * 128 + 4 * s + 2 * kh + 1];
  }
  for (int ct = 0; ct < 8; ++ct) {
    int n0 = ct * 16;
    v8f c = {0.f, 0.f, 0.f, 0.f, 0.f, 0.f, 0.f, 0.f};
#pragma unroll
    for (int s = 0; s < 32; ++s) {
      v2f a; a.x = areg[2 * s]; a.y = areg[2 * s + 1];
      int p = 2 * s + kh;  // K-pair index
      v2f b = *(const v2f*)&ldsW[(p * 128 + n0 + m) * 2];
      c = __builtin_amdgcn_wmma_f32_16x16x4_f32(false, a, false, b,
                                                (short)0, c, false, false);
    }
#pragma unroll
    for (int i = 0; i < 8; ++i)
      Hout[((long long)tile * 16 + i + 8 * kh) * 128 + n0 + m] = c[i];
  }
}

// ------------------------------------------------------------------
// GEMM2: G[N,64] = H2[N,32] * W2[32,64]   (f32 WMMA 16x16x4)
// ------------------------------------------------------------------
__global__ __launch_bounds__(256) void gemm2_kernel(
    const float* __restrict__ X, const float* __restrict__ W,
    float* __restrict__ Hout, int N) {
  __shared__ float ldsW[32 * 64];
  {
    unsigned base = (unsigned)(uintptr_t)ldsW;
    for (int i = threadIdx.x; i < 32 * 64 / 4; i += 256)
      async_g2l_b128(base + i * 16u, W + i * 4);
    wait_async0();
  }
  __syncthreads();
  {
    float tmp[8];
#pragma unroll
    for (int j = 0; j < 8; ++j) tmp[j] = ldsW[threadIdx.x * 8 + j];
    __syncthreads();
#pragma unroll
    for (int j = 0; j < 8; ++j) {
      int idx = threadIdx.x * 8 + j;
      int k = idx >> 6, n = idx & 63;
      ldsW[((k >> 1) * 64 + n) * 2 + (k & 1)] = tmp[j];
    }
  }
  __syncthreads();

  int wave = threadIdx.x >> 5;
  int lane = threadIdx.x & 31;
  int tile = blockIdx.x * 8 + wave;
  if (tile * 16 >= N) return;
  int m = lane & 15, kh = lane >> 4;
  long long row = (long long)tile * 16 + m;

  float areg[16];
#pragma unroll
  for (int s = 0; s < 8; ++s) {
    areg[2 * s]     = X[row * 32 + 4 * s + 2 * kh];
    areg[2 * s + 1] = X[row * 32 + 4 * s + 2 * kh + 1];
  }
  for (int ct = 0; ct < 4; ++ct) {
    int n0 = ct * 16;
    v8f c = {0.f, 0.f, 0.f, 0.f, 0.f, 0.f, 0.f, 0.f};
#pragma unroll
    for (int s = 0; s < 8; ++s) {
      v2f a; a.x = areg[2 * s]; a.y = areg[2 * s + 1];
      int p = 2 * s + kh;
      v2f b = *(const v2f*)&ldsW[(p * 64 + n0 + m) * 2];
      c = __builtin_amdgcn_wmma_f32_16x16x4_f32(false, a, false, b,
                                                (short)0, c, false, false);
    }
#pragma unroll
    for (int i = 0; i < 8; ++i)
      Hout[((long long)tile * 16 + i + 8 * kh) * 64 + n0 + m] = c[i];
  }
}

// per-(node,head) attention halves: as = <h[n,h,:], a_src[h,:]>, ad likewise
__global__ void attn_prep_kernel(const float* __restrict__ Hf,
                                 const float* __restrict__ a_src,
                                 const float* __restrict__ a_dst,
                                 float* __restrict__ as_, float* __restrict__ ad_,
                                 int N, int F) {
  int idx = blockIdx.x * blockDim.x + threadIdx.x;  // N*4
  if (idx >= N * 4) return;
  int n = idx >> 2, h = idx & 3;
  const float* hp = Hf + (long long)n * 4 * F + (long long)h * F;
  float s0 = 0.f, s1 = 0.f;
  for (int f = 0; f < F; ++f) {
    float v = hp[f];
    s0 += v * a_src[h * F + f];
    s1 += v * a_dst[h * F + f];
  }
  as_[idx] = s0;
  ad_[idx] = s1;
}

__device__ __forceinline__ float edge_e(const float* as_, const float* ad_,
                                        int src, int dst, int h) {
  float v = as_[src * 4 + h] + ad_[dst * 4 + h];
  return v > 0.f ? v : NEG_SLOPE * v;
}

__global__ void edge_max_kernel(const int* __restrict__ ei,
                                const float* __restrict__ as_,
                                const float* __restrict__ ad_,
                                unsigned* __restrict__ m_ord, int E) {
  int idx = blockIdx.x * blockDim.x + threadIdx.x;  // E*4
  if (idx >= E * 4) return;
  int e = idx >> 2, h = idx & 3;
  int src = ei[e], dst = ei[E + e];
  atomicMax(&m_ord[dst * 4 + h], f2ord(edge_e(as_, ad_, src, dst, h)));
}

__global__ void edge_sum_kernel(const int* __restrict__ ei,
                                const float* __restrict__ as_,
                                const float* __restrict__ ad_,
                                const unsigned* __restrict__ m_ord,
                                float* __restrict__ den, int E) {
  int idx = blockIdx.x * blockDim.x + threadIdx.x;
  if (idx >= E * 4) return;
  int e = idx >> 2, h = idx & 3;
  int src = ei[e], dst = ei[E + e];
  float v = edge_e(as_, ad_, src, dst, h);
  float mm = ord2f(m_ord[dst * 4 + h]);
  atomicAdd(&den[dst * 4 + h], expf(v - mm));
}

// scatter out[dst] += alpha * h[src]; one thread per (edge, head*feat)
__global__ void edge_agg_kernel(const int* __restrict__ ei,
                                const float* __restrict__ as_,
                                const float* __restrict__ ad_,
                                const unsigned* __restrict__ m_ord,
                                const float* __restrict__ den,
                                const float* __restrict__ Hf,
                                float* __restrict__ agg,
                                int E, int hfShift, int fShift) {
  long long idx = (long long)blockIdx.x * blockDim.x + threadIdx.x;
  long long total = (long long)E << hfShift;
  if (idx >= total) return;
  int e = (int)(idx >> hfShift);
  int hf = (int)(idx & ((1 << hfShift) - 1));
  int h = hf >> fShift;
  int src = ei[e], dst = ei[E + e];
  float v = edge_e(as_, ad_, src, dst, h);
  float mm = ord2f(m_ord[dst * 4 + h]);
  float alpha = expf(v - mm) / (den[dst * 4 + h] + EPSF);
  atomicAdd(&agg[((long long)dst << hfShift) + hf],
            alpha * Hf[((long long)src << hfShift) + hf]);
}

// head-mean + bias (+ optional relu)
__global__ void finalize_kernel(const float* __restrict__ agg,
                                const float* __restrict__ bias,
                                float* __restrict__ out, int N, int F, int relu) {
  int idx = blockIdx.x * blockDim.x + threadIdx.x;  // N*F
  if (idx >= N * F) return;
  int n = idx / F, f = idx % F;
  float s = 0.f;
  for (int h = 0; h < 4; ++h) s += agg[(long long)n * 4 * F + h * F + f];
  s = s * 0.25f + bias[f];
  if (relu) s = fmaxf(s, 0.f);
  out[idx] = s;
}

extern "C" void kernel_launch(void* const* d_in, const int* in_sizes, int n_in,
                              void* d_out, int out_size, void* d_ws, size_t ws_size,
                              hipStream_t stream) {
  const float* x   = (const float*)d_in[0];
  const int*   ei  = (const int*)d_in[1];
  const float* W1  = (const float*)d_in[2];
  const float* a1s = (const float*)d_in[3];
  const float* a1d = (const float*)d_in[4];
  const float* b1  = (const float*)d_in[5];
  const float* W2  = (const float*)d_in[6];
  const float* a2s = (const float*)d_in[7];
  const float* a2d = (const float*)d_in[8];
  const float* b2  = (const float*)d_in[9];

  int N = in_sizes[0] / 128;
  int E = in_sizes[1] / 2;

  // ---- workspace layout (floats); layer-2 g2/agg2 alias h1 region ----
  float* ws   = (float*)d_ws;
  float* h1   = ws;                               // N*128
  float* agg1 = h1 + (size_t)N * 128;             // N*128
  float* h2   = agg1 + (size_t)N * 128;           // N*32
  float* as1  = h2 + (size_t)N * 32;              // N*4 each below
  float* ad1  = as1 + (size_t)N * 4;
  unsigned* m1 = (unsigned*)(ad1 + (size_t)N * 4);
  float* den1 = (float*)(m1 + (size_t)N * 4);
  float* as2  = den1 + (size_t)N * 4;
  float* ad2  = as2 + (size_t)N * 4;
  unsigned* m2 = (unsigned*)(ad2 + (size_t)N * 4);
  float* den2 = (float*)(m2 + (size_t)N * 4);
  float* g2   = h1;                               // reuse: N*64
  float* agg2 = h1 + (size_t)N * 64;              // reuse: N*64

  int tiles = (N + 15) / 16;
  int gGemm = (tiles + 7) / 8;
  int tEH = E * 4;

  // ---- layer 1 ----
  zero_kernel<<<2048, 256, 0, stream>>>(agg1, (long long)N * 128);
  zero_kernel<<<512, 256, 0, stream>>>((float*)m1, (long long)N * 8);  // m1+den1
  gemm1_kernel<<<gGemm, 256, 0, stream>>>(x, W1, h1, N);
  attn_prep_kernel<<<(N * 4 + 255) / 256, 256, 0, stream>>>(h1, a1s, a1d, as1, ad1, N, 32);
  edge_max_kernel<<<(tEH + 255) / 256, 256, 0, stream>>>(ei, as1, ad1, m1, E);
  edge_sum_kernel<<<(tEH + 255) / 256, 256, 0, stream>>>(ei, as1, ad1, m1, den1, E);
  {
    long long tot = (long long)E * 128;
    edge_agg_kernel<<<(unsigned)((tot + 255) / 256), 256, 0, stream>>>(
        ei, as1, ad1, m1, den1, h1, agg1, E, 7, 5);
  }
  finalize_kernel<<<(N * 32 + 255) / 256, 256, 0, stream>>>(agg1, b1, h2, N, 32, 1);

  // ---- layer 2 ----
  zero_kernel<<<1024, 256, 0, stream>>>(agg2, (long long)N * 64);
  zero_kernel<<<512, 256, 0, stream>>>((float*)m2, (long long)N * 8);  // m2+den2
  gemm2_kernel<<<gGemm, 256, 0, stream>>>(h2, W2, g2, N);
  attn_prep_kernel<<<(N * 4 + 255) / 256, 256, 0, stream>>>(g2, a2s, a2d, as2, ad2, N, 16);
  edge_max_kernel<<<(tEH + 255) / 256, 256, 0, stream>>>(ei, as2, ad2, m2, E);
  edge_sum_kernel<<<(tEH + 255) / 256, 256, 0, stream>>>(ei, as2, ad2, m2, den2, E);
  {
    long long tot = (long long)E * 64;
    edge_agg_kernel<<<(unsigned)((tot + 255) / 256), 256, 0, stream>>>(
        ei, as2, ad2, m2, den2, g2, agg2, E, 6, 4);
  }
  finalize_kernel<<<(N * 16 + 255) / 256, 256, 0, stream>>>(agg2, b2, (float*)d_out, N, 16, 0);
}